// GCN_Encoder_45947560132669
// MI455X (gfx1250) — compile-verified
//
#include <hip/hip_runtime.h>
#include <hip/hip_bf16.h>

typedef __attribute__((ext_vector_type(2))) float v2f;
typedef __attribute__((ext_vector_type(8))) float v8f;

#define D_FEAT 128
#define LDS_A_STRIDE 132   // padded: row*132 % 64 banks => 4-bank step per row, conflict-free halves

// ---------------- utility: zero a float buffer ----------------
__global__ void gcn_zero_kernel(float* __restrict__ p, size_t n) {
    size_t i = (size_t)blockIdx.x * blockDim.x + threadIdx.x;
    if (i < n) p[i] = 0.0f;
}

// ---------------- degrees via edge-parallel atomics ----------------
__global__ void gcn_degree_kernel(const long long* __restrict__ src,
                                  const long long* __restrict__ dst,
                                  float* __restrict__ deg_out,
                                  float* __restrict__ deg_in, int nedges) {
    int e = blockIdx.x * blockDim.x + threadIdx.x;
    if (e >= nedges) return;
    atomicAdd(&deg_out[src[e]], 1.0f);
    atomicAdd(&deg_in[dst[e]], 1.0f);
}

// norm = rsqrt(max(deg,1)) in place
__global__ void gcn_norm_kernel(float* __restrict__ deg, int n) {
    int i = blockIdx.x * blockDim.x + threadIdx.x;
    if (i >= n) return;
    float d = deg[i];
    d = d < 1.0f ? 1.0f : d;
    deg[i] = rsqrtf(d);
}

// ---------------- gather + scatter-add: agg[dst] += h[src] * norm_src[src] ----
// one thread per (edge, 4 consecutive features): 32 threads per edge
__global__ void gcn_scatter_kernel(const float* __restrict__ h,
                                   const float* __restrict__ norm_src,
                                   const long long* __restrict__ src,
                                   const long long* __restrict__ dst,
                                   float* __restrict__ agg, int nedges) {
    int t = blockIdx.x * blockDim.x + threadIdx.x;
    int e = t >> 5;
    if (e >= nedges) return;
    int c = (t & 31) * 4;
    long long s = src[e];
    long long d = dst[e];
    float ns = norm_src[s];
    const float4 hv = *(const float4*)(h + (size_t)s * D_FEAT + c);
    float* ap = agg + (size_t)d * D_FEAT + c;
    atomicAdd(ap + 0, hv.x * ns);
    atomicAdd(ap + 1, hv.y * ns);
    atomicAdd(ap + 2, hv.z * ns);
    atomicAdd(ap + 3, hv.w * ns);
}

// ---------------- fused (agg*norm_dst) @ W + b -> relu -> + res_scale*resid ----
// block = 256 threads = 8 waves. Block owns 16 rows (m-tile), wave w owns cols [16w,16w+16).
// WMMA f32 16x16x4, K-loop over D=128 in steps of 4.
__global__ __launch_bounds__(256) void gcn_gemm_kernel(
    const float* __restrict__ A,        // agg [N][128]
    const float* __restrict__ norm_dst, // [N]
    const float* __restrict__ W,        // [128][128] (in,out)
    const float* __restrict__ bias,     // [128]
    const float* __restrict__ resid,    // features [N][128]
    float res_scale,
    float* __restrict__ out)            // [N][128]
{
    __shared__ float Atile[16 * LDS_A_STRIDE];

    const int m0 = blockIdx.x * 16;
    const int tid = threadIdx.x;

    // Cooperative load of A-tile into LDS, scaled by norm_dst (8 floats / thread).
    {
        int r = tid >> 4;          // 0..15 row within tile
        int c = (tid & 15) * 8;    // 0..120 col start
        float nd = norm_dst[m0 + r];
        const float* gsrc = A + (size_t)(m0 + r) * D_FEAT + c;
        float* ldst = Atile + r * LDS_A_STRIDE + c;
#pragma unroll
        for (int i = 0; i < 8; ++i) ldst[i] = gsrc[i] * nd;
    }
    __syncthreads();

    const int wid  = tid >> 5;   // wave id 0..7 -> output col tile
    const int lane = tid & 31;
    const int hi   = lane >> 4;  // lane half selects K offset {0,2}
    const int ml   = lane & 15;  // A: M row / B: N col within tile
    const int n0   = wid * 16;
    const int col  = n0 + ml;

    v8f acc = {};
    const float* arow = Atile + ml * LDS_A_STRIDE;

#pragma unroll
    for (int k = 0; k < D_FEAT / 4; ++k) {
        const int kk = k * 4 + 2 * hi;
        v2f a;
        a.x = arow[kk];
        a.y = arow[kk + 1];
        v2f b;
        b.x = W[(size_t)kk * D_FEAT + col];
        b.y = W[(size_t)(kk + 1) * D_FEAT + col];
        // 8 args: (neg_a, A, neg_b, B, c_mod, C, reuse_a, reuse_b)
        acc = __builtin_amdgcn_wmma_f32_16x16x4_f32(
            false, a, false, b, (short)0, acc, false, false);
    }

    // D layout: VGPR r -> row = m0 + r + 8*hi, col = n0 + (lane&15)
    const float bv = bias[col];
#pragma unroll
    for (int r = 0; r < 8; ++r) {
        const int row = m0 + r + 8 * hi;
        float v = acc[r] + bv;
        v = v > 0.0f ? v : 0.0f;
        const size_t idx = (size_t)row * D_FEAT + col;
        out[idx] = v + res_scale * resid[idx];
    }
}

extern "C" void kernel_launch(void* const* d_in, const int* in_sizes, int n_in,
                              void* d_out, int out_size, void* d_ws, size_t ws_size,
                              hipStream_t stream) {
    const float*     features = (const float*)d_in[0];
    const long long* src      = (const long long*)d_in[1];
    const long long* dst      = (const long long*)d_in[2];
    const float*     W1       = (const float*)d_in[3];
    const float*     b1       = (const float*)d_in[4];
    const float*     W2       = (const float*)d_in[5];
    const float*     b2       = (const float*)d_in[6];
    float* out = (float*)d_out;

    const int N = in_sizes[0] / D_FEAT;   // 50000
    const int E = in_sizes[1];            // 600000

    // workspace layout (floats): [norm_src N][norm_dst N][agg N*D][h1 N*D]
    float* nsrc = (float*)d_ws;
    float* ndst = nsrc + N;
    float* agg  = ndst + N;
    float* h1   = agg + (size_t)N * D_FEAT;

    const int TPB = 256;
    const size_t twoN = 2 * (size_t)N;
    const size_t ND   = (size_t)N * D_FEAT;

    // degrees -> norms
    gcn_zero_kernel<<<(int)((twoN + TPB - 1) / TPB), TPB, 0, stream>>>(nsrc, twoN);
    gcn_degree_kernel<<<(E + TPB - 1) / TPB, TPB, 0, stream>>>(src, dst, nsrc, ndst, E);
    gcn_norm_kernel<<<(int)((twoN + TPB - 1) / TPB), TPB, 0, stream>>>(nsrc, (int)twoN);

    const int scatter_blocks = (int)(((size_t)E * 32 + TPB - 1) / TPB);
    const int gemm_blocks = N / 16;  // 50000 / 16 = 3125 exact
    const int zero_nd_blocks = (int)((ND + TPB - 1) / TPB);

    // ---- layer 1: h1 = relu((scatter(features)) @ W1 + b1) + features
    gcn_zero_kernel<<<zero_nd_blocks, TPB, 0, stream>>>(agg, ND);
    gcn_scatter_kernel<<<scatter_blocks, TPB, 0, stream>>>(features, nsrc, src, dst, agg, E);
    gcn_gemm_kernel<<<gemm_blocks, TPB, 0, stream>>>(agg, ndst, W1, b1, features, 1.0f, h1);

    // ---- layer 2 + final: out = relu((scatter(h1)) @ W2 + b2) + 2*features
    gcn_zero_kernel<<<zero_nd_blocks, TPB, 0, stream>>>(agg, ND);
    gcn_scatter_kernel<<<scatter_blocks, TPB, 0, stream>>>(h1, nsrc, src, dst, agg, E);
    gcn_gemm_kernel<<<gemm_blocks, TPB, 0, stream>>>(agg, ndst, W2, b2, features, 2.0f, out);
}